// Model_87857851007503
// MI455X (gfx1250) — compile-verified
//
#include <hip/hip_runtime.h>

// Problem constants (from reference)
#define B_ 16
#define L_ 4096
#define D_ 512
#define S_ 32
#define M_ 64
#define E_ 16
#define P_ 240
#define DIS_ 20
#define FCH_ 1024
#define R_ 97
#define KX_ (3 * D_ + DIS_) /* 1556 */
#define MT32_ (B_ * P_ / 32) /* 120 m-tiles of 32 rows */

typedef float v2f __attribute__((ext_vector_type(2)));
typedef float v8f __attribute__((ext_vector_type(8)));

// Order-preserving float -> u32 key (for max via integer atomics).
__device__ __forceinline__ unsigned fkey(float f) {
  unsigned u = __float_as_uint(f);
  return (u & 0x80000000u) ? ~u : (u | 0x80000000u);
}
__device__ __forceinline__ float fdec(unsigned k) {
  unsigned u = (k & 0x80000000u) ? (k & 0x7fffffffu) : ~k;
  return __uint_as_float(u);
}
#define NEG_INF_KEY 0x007FFFFFu /* key of -inf */

__global__ void init_keys(unsigned* __restrict__ p, int n) {
  int i = blockIdx.x * blockDim.x + threadIdx.x;
  if (i < n) p[i] = NEG_INF_KEY;
}

// Segment-max of encoder_outputs into sentence + mention keys.
// grid (D/64, 8 L-splits, B), block 256 = 64 dims x 4 token lanes.
__global__ __launch_bounds__(256) void pool_kernel(
    const float* __restrict__ enc, const int* __restrict__ sid,
    const int* __restrict__ mid, unsigned* __restrict__ sentk,
    unsigned* __restrict__ menk) {
  __shared__ unsigned u[(S_ + M_) * 64];
  const int tid = threadIdx.x;
  for (int i = tid; i < (S_ + M_) * 64; i += 256) u[i] = NEG_INF_KEY;
  __syncthreads();
  const int tx = tid & 63, ty = tid >> 6;
  const int d0 = blockIdx.x * 64;
  const int b = blockIdx.z;
  const int t0 = blockIdx.y * (L_ / 8);
  for (int t = t0 + ty; t < t0 + L_ / 8; t += 4) {
    int s = sid[b * L_ + t];
    int m = mid[b * L_ + t];
    float v = enc[((size_t)b * L_ + t) * D_ + d0 + tx];
    unsigned k = fkey(v);
    atomicMax(&u[s * 64 + tx], k);          // ds_max_u32
    atomicMax(&u[(S_ + m) * 64 + tx], k);   // ds_max_u32
  }
  __syncthreads();
  for (int i = tid; i < (S_ + M_) * 64; i += 256) {
    int seg = i >> 6, dd = i & 63;
    unsigned k = u[i];
    if (k != NEG_INF_KEY) {
      if (seg < S_)
        atomicMax(&sentk[((size_t)(b * S_ + seg)) * D_ + d0 + dd], k);
      else
        atomicMax(&menk[((size_t)(b * M_ + (seg - S_))) * D_ + d0 + dd], k);
    }
  }
}

// Per batch: doc = max over sentences; ent = segment-max of mentions.
__global__ __launch_bounds__(256) void reduce_kernel(
    const unsigned* __restrict__ sentk, const unsigned* __restrict__ menk,
    const int* __restrict__ mids, float* __restrict__ ent,
    float* __restrict__ doc) {
  __shared__ float entl[E_ * D_];
  __shared__ int idsl[M_];
  const int tid = threadIdx.x, b = blockIdx.x;
  if (tid < M_) idsl[tid] = mids[b * M_ + tid];
  for (int c = 0; c < 2; ++c) {
    int d = tid + c * 256;
    for (int e = 0; e < E_; ++e) entl[e * D_ + d] = -__builtin_inff();
  }
  __syncthreads();
  for (int c = 0; c < 2; ++c) {
    int d = tid + c * 256;
    float mx = -__builtin_inff();
    for (int s = 0; s < S_; ++s)
      mx = fmaxf(mx, fdec(sentk[((size_t)(b * S_ + s)) * D_ + d]));
    doc[b * D_ + d] = mx;
    for (int m = 0; m < M_; ++m) {
      float v = fdec(menk[((size_t)(b * M_ + m)) * D_ + d]);
      int e = idsl[m];  // column d owned by this thread: no race
      entl[e * D_ + d] = fmaxf(entl[e * D_ + d], v);
    }
    for (int e = 0; e < E_; ++e)
      ent[((size_t)(b * E_ + e)) * D_ + d] = entl[e * D_ + d];
  }
}

// Fill X columns [512,1556): doc broadcast, context=max(head,tail), dis embed.
__global__ __launch_bounds__(256) void gather_kernel(
    const float* __restrict__ ent, const float* __restrict__ doc,
    const int* __restrict__ head, const int* __restrict__ tail,
    const int* __restrict__ dis, const float* __restrict__ dtab,
    float* __restrict__ X) {
  const int row = blockIdx.x;
  const int b = row / P_;
  const int h = head[row], t = tail[row], dd = dis[row];
  const float* hp = ent + ((size_t)(b * E_ + h)) * D_;
  const float* tp = ent + ((size_t)(b * E_ + t)) * D_;
  const float* dp = doc + (size_t)b * D_;
  float* xr = X + (size_t)row * KX_;
  for (int d = threadIdx.x; d < D_; d += 256) {
    xr[D_ + d] = dp[d];
    xr[2 * D_ + d] = fmaxf(hp[d], tp[d]);
  }
  if (threadIdx.x < DIS_)
    xr[3 * D_ + threadIdx.x] = dtab[dd * DIS_ + threadIdx.x];
}

// Inner 512-wide K panel: 8 WMMAs per k-step from 2 A b64 + 8 B b32 loads.
__device__ __forceinline__ void mma_panel_512(
    const float* __restrict__ a0base, const float* __restrict__ a1base,
    const float* __restrict__ Bw, int kofs, int n0, int lm, int kh,
    v8f acc0[4], v8f acc1[4]) {
  for (int k = 0; k < D_; k += 4) {
    v2f a0 = *(const v2f*)(a0base + k + kh);
    v2f a1 = *(const v2f*)(a1base + k + kh);
#pragma unroll
    for (int j = 0; j < 4; ++j) {
      int n = n0 + j * 16 + lm;
      v2f bf;
      bf.x = Bw[(size_t)(kofs + k + kh) * D_ + n];
      bf.y = Bw[(size_t)(kofs + k + kh + 1) * D_ + n];
      acc0[j] = __builtin_amdgcn_wmma_f32_16x16x4_f32(
          false, a0, false, bf, (short)0, acc0[j], false, false);
      acc1[j] = __builtin_amdgcn_wmma_f32_16x16x4_f32(
          false, a1, false, bf, (short)0, acc1[j], false, false);
    }
  }
}

// Wc GEMM with fused head/tail gather for A rows.
// Each wave: 32(M) x 64(N) tile of X[:,0:512]; K = 1024 split into head
// (k<512) and tail (k>=512) panels so base pointers are loop-invariant.
__global__ __launch_bounds__(256) void gemm_pair_kernel(
    const float* __restrict__ ent, const int* __restrict__ head,
    const int* __restrict__ tail, const float* __restrict__ Wc,
    const float* __restrict__ Wcb, float* __restrict__ X) {
  const int wave = (blockIdx.x * 256 + threadIdx.x) >> 5;
  const int lane = threadIdx.x & 31;
  const int mt = wave % MT32_;
  const int n0 = (wave / MT32_) * 64;
  const int lm = lane & 15;
  const int kh = (lane >> 4) << 1;  // frag half: K+0/1 (lanes 0-15) vs K+2/3
  const int m0 = mt * 32 + lm;      // rows m0 and m0+16 per lane
  const int m1 = m0 + 16;
  const int b0 = m0 / P_, b1 = m1 / P_;
  const float* h0 = ent + ((size_t)(b0 * E_ + head[m0])) * D_;
  const float* t0p = ent + ((size_t)(b0 * E_ + tail[m0])) * D_;
  const float* h1 = ent + ((size_t)(b1 * E_ + head[m1])) * D_;
  const float* t1p = ent + ((size_t)(b1 * E_ + tail[m1])) * D_;
  v8f acc0[4] = {};
  v8f acc1[4] = {};
  mma_panel_512(h0, h1, Wc, 0, n0, lm, kh, acc0, acc1);     // k in [0,512)
  mma_panel_512(t0p, t1p, Wc, D_, n0, lm, kh, acc0, acc1);  // k in [512,1024)
  const int rbase = mt * 32 + (lane >> 4) * 8;
#pragma unroll
  for (int j = 0; j < 4; ++j) {
    int n = n0 + j * 16 + lm;
    float bv = Wcb[n];
#pragma unroll
    for (int i = 0; i < 8; ++i) {
      X[(size_t)(rbase + i) * KX_ + n] = acc0[j][i] + bv;
      X[(size_t)(rbase + 16 + i) * KX_ + n] = acc1[j][i] + bv;
    }
  }
}

// Generic fp32 WMMA GEMM: C[M x ncols] = relu?(A[M x K] * Bw[K x ncols] + bias)
// M = 3840; one wave per 32x64 tile; grid waves = MT32_ * ceil(N/64).
// BOUND=false strips per-load column bounds checks (N multiple of 64).
template <bool BOUND, bool RELU>
__global__ __launch_bounds__(256) void gemm_kernel(
    const float* __restrict__ A, int lda, const float* __restrict__ Bw,
    int ldb, int ncols, const float* __restrict__ bias, float* __restrict__ C,
    int ldc, int K) {
  const int wave = (blockIdx.x * 256 + threadIdx.x) >> 5;
  const int lane = threadIdx.x & 31;
  const int mt = wave % MT32_;
  const int n0 = (wave / MT32_) * 64;
  const int lm = lane & 15;
  const int kh = (lane >> 4) << 1;
  const float* arow0 = A + (size_t)(mt * 32 + lm) * lda;
  const float* arow1 = arow0 + (size_t)16 * lda;
  v8f acc0[4] = {};
  v8f acc1[4] = {};
  for (int k = 0; k < K; k += 4) {
    v2f a0 = *(const v2f*)(arow0 + k + kh);
    v2f a1 = *(const v2f*)(arow1 + k + kh);
#pragma unroll
    for (int j = 0; j < 4; ++j) {
      int n = n0 + j * 16 + lm;
      v2f bf;
      if (BOUND) {
        bf.x = (n < ncols) ? Bw[(size_t)(k + kh) * ldb + n] : 0.f;
        bf.y = (n < ncols) ? Bw[(size_t)(k + kh + 1) * ldb + n] : 0.f;
      } else {
        bf.x = Bw[(size_t)(k + kh) * ldb + n];
        bf.y = Bw[(size_t)(k + kh + 1) * ldb + n];
      }
      acc0[j] = __builtin_amdgcn_wmma_f32_16x16x4_f32(
          false, a0, false, bf, (short)0, acc0[j], false, false);
      acc1[j] = __builtin_amdgcn_wmma_f32_16x16x4_f32(
          false, a1, false, bf, (short)0, acc1[j], false, false);
    }
  }
  const int rbase = mt * 32 + (lane >> 4) * 8;
#pragma unroll
  for (int j = 0; j < 4; ++j) {
    int n = n0 + j * 16 + lm;
    if (!BOUND || n < ncols) {
      float bv = bias[n];
#pragma unroll
      for (int i = 0; i < 8; ++i) {
        float v0 = acc0[j][i] + bv;
        float v1 = acc1[j][i] + bv;
        if (RELU) {
          v0 = fmaxf(v0, 0.f);
          v1 = fmaxf(v1, 0.f);
        }
        C[(size_t)(rbase + i) * ldc + n] = v0;
        C[(size_t)(rbase + 16 + i) * ldc + n] = v1;
      }
    }
  }
}

extern "C" void kernel_launch(void* const* d_in, const int* in_sizes, int n_in,
                              void* d_out, int out_size, void* d_ws,
                              size_t ws_size, hipStream_t stream) {
  const float* enc = (const float*)d_in[0];
  const int* sid = (const int*)d_in[1];
  const int* mid = (const int*)d_in[2];
  const int* ment = (const int*)d_in[3];
  const int* head = (const int*)d_in[4];
  const int* tail = (const int*)d_in[5];
  const int* dis = (const int*)d_in[6];
  const float* dtab = (const float*)d_in[7];
  const float* Wc_w = (const float*)d_in[8];
  const float* Wc_b = (const float*)d_in[9];
  const float* fc1_w = (const float*)d_in[10];
  const float* fc1_b = (const float*)d_in[11];
  const float* fc2_w = (const float*)d_in[12];
  const float* fc2_b = (const float*)d_in[13];

  // Workspace layout (total ~43.3 MB)
  char* ws = (char*)d_ws;
  unsigned* sentk = (unsigned*)ws;                                   // 1.0 MB
  unsigned* menk = (unsigned*)(ws + (size_t)B_ * S_ * D_ * 4);       // 2.0 MB
  float* ent = (float*)(ws + (size_t)B_ * (S_ + M_) * D_ * 4);       // 0.5 MB
  float* doc = (float*)((char*)ent + (size_t)B_ * E_ * D_ * 4);      // 32 KB
  float* X = (float*)((char*)doc + (size_t)B_ * D_ * 4);             // 23.9 MB
  float* H = (float*)((char*)X + (size_t)B_ * P_ * KX_ * 4);         // 15.7 MB
  float* out = (float*)d_out;

  const int nkeys = B_ * (S_ + M_) * D_;
  init_keys<<<(nkeys + 255) / 256, 256, 0, stream>>>(sentk, nkeys);
  pool_kernel<<<dim3(D_ / 64, 8, B_), 256, 0, stream>>>(enc, sid, mid, sentk,
                                                        menk);
  reduce_kernel<<<B_, 256, 0, stream>>>(sentk, menk, ment, ent, doc);
  gather_kernel<<<B_ * P_, 256, 0, stream>>>(ent, doc, head, tail, dis, dtab,
                                             X);
  // Wc: 120 mtiles x 8 n-tiles = 960 waves -> 120 blocks
  gemm_pair_kernel<<<MT32_ * (D_ / 64) / 8, 256, 0, stream>>>(
      ent, head, tail, Wc_w, Wc_b, X);
  // fc1: 120 x 16 = 1920 waves -> 240 blocks (no bounds checks, relu)
  gemm_kernel<false, true><<<MT32_ * (FCH_ / 64) / 8, 256, 0, stream>>>(
      X, KX_, fc1_w, FCH_, FCH_, fc1_b, H, FCH_, KX_);
  // fc2: 120 x 2 = 240 waves -> 30 blocks (N=97 bounds-checked)
  gemm_kernel<true, false><<<MT32_ * 2 / 8, 256, 0, stream>>>(
      H, FCH_, fc2_w, R_, R_, fc2_b, out, R_, FCH_);
}